// CoreCapsuleNetwork_42090679501341
// MI455X (gfx1250) — compile-verified
//
#include <hip/hip_runtime.h>
#include <hip/hip_bf16.h>

// CoreCapsuleNetwork dynamic routing for MI455X (gfx1250, wave32).
// B=4096, SEQ=50, K=4, H=64, 3 routing iterations.
// Memory-bound: 3 streams of the 210MB item_eb_hat tensor => ~27us floor at 23.3TB/s.
// Per-(b,k) matmuls are mapped onto V_WMMA_F32_16X16X4_F32 (fp32, exact layout per ISA 7.12.2).

#define BATCH 4096
#define SEQL  50
#define SEQP  64      // padded seq (zero rows 50..63)
#define KNUM  4
#define HID   64
#define COLS  (KNUM * SEQL)   // 200
#define XROWSTRIDE 68         // LDS row stride in floats: 16B-aligned, bank-conflict free

typedef __attribute__((ext_vector_type(2))) float v2f;
typedef __attribute__((ext_vector_type(8))) float v8f;

// ---------------------------------------------------------------------------
// Kernel 0: copy capsule_weight into workspace (inputs are immutable)
// ---------------------------------------------------------------------------
__global__ void cw_copy_kernel(const float* __restrict__ src, float* __restrict__ dst, int n) {
    int i = blockIdx.x * blockDim.x + threadIdx.x;
    if (i < n) dst[i] = src[i];
}

// ---------------------------------------------------------------------------
// Kernel 1: batch-axis softmax stats. One block per column (k*SEQ+s):
// max_b cw[b,col] and sum_b exp(cw[b,col]-max). 200 columns x 4096 rows.
// ---------------------------------------------------------------------------
__global__ void col_stats_kernel(const float* __restrict__ cw,
                                 float* __restrict__ colmax,
                                 float* __restrict__ colsum) {
    __shared__ float red[256];
    const int col = blockIdx.x;     // 0..COLS-1
    const int t   = threadIdx.x;    // 0..255

    float m = -INFINITY;
    for (int b = t; b < BATCH; b += 256)
        m = fmaxf(m, cw[(size_t)b * COLS + col]);
    red[t] = m;
    __syncthreads();
    for (int off = 128; off > 0; off >>= 1) {
        if (t < off) red[t] = fmaxf(red[t], red[t + off]);
        __syncthreads();
    }
    const float cmax = red[0];
    __syncthreads();

    float s = 0.f;
    for (int b = t; b < BATCH; b += 256)
        s += __expf(cw[(size_t)b * COLS + col] - cmax);
    red[t] = s;
    __syncthreads();
    for (int off = 128; off > 0; off >>= 1) {
        if (t < off) red[t] += red[t + off];
        __syncthreads();
    }
    if (t == 0) {
        colmax[col] = cmax;
        colsum[col] = red[0];
    }
}

// ---------------------------------------------------------------------------
// Kernel 2/3: one wave (32 lanes) per (b,k).
//   FINAL=false: v = w*X ; squash scalar ; cw += scalar * (X*v)
//   FINAL=true : v = w*X ; squash scalar ; out = scalar * v
// Both products via v_wmma_f32_16x16x4_f32.
// ---------------------------------------------------------------------------
template <bool FINAL>
__global__ void route_iter_kernel(const float* __restrict__ x,       // [B,SEQ,K,H]
                                  const int*   __restrict__ mask,    // [B,SEQ]
                                  float*       __restrict__ cw,      // [B,K,SEQ] (workspace)
                                  const float* __restrict__ colmax,  // [COLS]
                                  const float* __restrict__ colsum,  // [COLS]
                                  float*       __restrict__ out) {   // [B,K,H]
    __shared__ __align__(16) float Xs[SEQP * XROWSTRIDE];  // X[b,k] padded to 64 rows
    __shared__ float wbuf[SEQP];
    __shared__ float vbuf[HID];
    __shared__ float dbuf[SEQP];

    const int bk   = blockIdx.x;        // 0..B*K-1
    const int b    = bk >> 2;
    const int k    = bk & 3;
    const int lane = threadIdx.x;       // 0..31
    const int l16  = lane & 15;
    const bool hi  = lane >= 16;

    // ---- Stream X[b, :, k, :] into LDS (row s at Xs[s*XROWSTRIDE]) ----
    // global element offset: b*SEQ*K*H + s*K*H + k*H = b*12800 + s*256 + k*64
    const float* gbase = x + (size_t)b * (SEQL * KNUM * HID) + (size_t)k * HID;
    for (int idx = lane; idx < SEQL * 16; idx += 32) {      // 50 rows x 16 float4
        const int s  = idx >> 4;
        const int c4 = (idx & 15) << 2;
        float4 v = *(const float4*)(gbase + (size_t)s * (KNUM * HID) + c4);
        *(float4*)&Xs[s * XROWSTRIDE + c4] = v;
    }
    for (int idx = lane; idx < (SEQP - SEQL) * 16; idx += 32) {  // zero pad rows 50..63
        const int s  = SEQL + (idx >> 4);
        const int c4 = (idx & 15) << 2;
        *(float4*)&Xs[s * XROWSTRIDE + c4] = make_float4(0.f, 0.f, 0.f, 0.f);
    }

    // ---- masked softmax weights w[s] (zero-padded to 64) ----
    for (int s = lane; s < SEQP; s += 32) {
        float wv = 0.f;
        if (s < SEQL) {
            const int col = k * SEQL + s;
            const float e = __expf(cw[(size_t)b * COLS + col] - colmax[col]) / colsum[col];
            wv = (mask[(size_t)b * SEQL + s] == 0) ? 0.f : e;
        }
        wbuf[s] = wv;
    }
    __syncthreads();

    // ---- Matmul 1: v(1x64) = w(1x64) * X(64x64) via WMMA 16x16x4 f32 ----
    // A: row 0 = w chunk (lane0: K=0,1 ; lane16: K=2,3), rows 1..15 zero.
    // B: X[k0..k0+3][nt*16 .. nt*16+15]
    v8f acc0 = {}, acc1 = {}, acc2 = {}, acc3 = {};
#pragma unroll
    for (int kt = 0; kt < SEQP / 4; ++kt) {
        const int k0 = kt * 4;
        v2f a;
        a.x = (lane == 0) ? wbuf[k0]     : ((lane == 16) ? wbuf[k0 + 2] : 0.f);
        a.y = (lane == 0) ? wbuf[k0 + 1] : ((lane == 16) ? wbuf[k0 + 3] : 0.f);
        const int krow = k0 + (hi ? 2 : 0);
        const float* r0 = &Xs[krow * XROWSTRIDE];
        const float* r1 = &Xs[(krow + 1) * XROWSTRIDE];
        v2f b0; b0.x = r0[ 0 + l16]; b0.y = r1[ 0 + l16];
        v2f b1; b1.x = r0[16 + l16]; b1.y = r1[16 + l16];
        v2f b2; b2.x = r0[32 + l16]; b2.y = r1[32 + l16];
        v2f b3; b3.x = r0[48 + l16]; b3.y = r1[48 + l16];
        acc0 = __builtin_amdgcn_wmma_f32_16x16x4_f32(false, a, false, b0, (short)0, acc0, false, false);
        acc1 = __builtin_amdgcn_wmma_f32_16x16x4_f32(false, a, false, b1, (short)0, acc1, false, false);
        acc2 = __builtin_amdgcn_wmma_f32_16x16x4_f32(false, a, false, b2, (short)0, acc2, false, false);
        acc3 = __builtin_amdgcn_wmma_f32_16x16x4_f32(false, a, false, b3, (short)0, acc3, false, false);
    }
    // D row M=0 lives in VGPR0, lanes 0..15 (N = lane)
    if (lane < 16) {
        vbuf[ 0 + lane] = acc0[0];
        vbuf[16 + lane] = acc1[0];
        vbuf[32 + lane] = acc2[0];
        vbuf[48 + lane] = acc3[0];
    }
    __syncthreads();

    // ---- squash scalar: n = |v|^2 ; scalar = n/(1+n)/sqrt(n+1e-9) ----
    float p = vbuf[lane] * vbuf[lane] + vbuf[lane + 32] * vbuf[lane + 32];
#pragma unroll
    for (int off = 16; off > 0; off >>= 1) p += __shfl_xor(p, off, 32);
    const float n = p;
    const float scalar = (n / (1.f + n)) * rsqrtf(n + 1e-9f);

    if constexpr (FINAL) {
        float* o = out + (size_t)bk * HID;
        o[lane]      = scalar * vbuf[lane];
        o[lane + 32] = scalar * vbuf[lane + 32];
    } else {
        // ---- Matmul 2: delta(64x1) = X(64x64) * v(64x1) via WMMA 16x16x4 f32 ----
        // B: column 0 = v chunk (lane0: K=0,1 ; lane16: K=2,3); A: X row tiles.
        v8f d0 = {}, d1 = {}, d2 = {}, d3 = {};
#pragma unroll
        for (int kt = 0; kt < HID / 4; ++kt) {
            const int k0 = kt * 4;
            v2f bf;
            bf.x = (lane == 0) ? vbuf[k0]     : ((lane == 16) ? vbuf[k0 + 2] : 0.f);
            bf.y = (lane == 0) ? vbuf[k0 + 1] : ((lane == 16) ? vbuf[k0 + 3] : 0.f);
            const int kk = k0 + (hi ? 2 : 0);
            v2f a0; a0.x = Xs[( 0 + l16) * XROWSTRIDE + kk]; a0.y = Xs[( 0 + l16) * XROWSTRIDE + kk + 1];
            v2f a1; a1.x = Xs[(16 + l16) * XROWSTRIDE + kk]; a1.y = Xs[(16 + l16) * XROWSTRIDE + kk + 1];
            v2f a2; a2.x = Xs[(32 + l16) * XROWSTRIDE + kk]; a2.y = Xs[(32 + l16) * XROWSTRIDE + kk + 1];
            v2f a3; a3.x = Xs[(48 + l16) * XROWSTRIDE + kk]; a3.y = Xs[(48 + l16) * XROWSTRIDE + kk + 1];
            d0 = __builtin_amdgcn_wmma_f32_16x16x4_f32(false, a0, false, bf, (short)0, d0, false, false);
            d1 = __builtin_amdgcn_wmma_f32_16x16x4_f32(false, a1, false, bf, (short)0, d1, false, false);
            d2 = __builtin_amdgcn_wmma_f32_16x16x4_f32(false, a2, false, bf, (short)0, d2, false, false);
            d3 = __builtin_amdgcn_wmma_f32_16x16x4_f32(false, a3, false, bf, (short)0, d3, false, false);
        }
        // Column N=0: lane0 holds M = mt*16 + i (VGPR i), lane16 holds M = mt*16 + 8 + i
        if (lane == 0 || lane == 16) {
            const int mb = hi ? 8 : 0;
#pragma unroll
            for (int i = 0; i < 8; ++i) {
                dbuf[ 0 + mb + i] = d0[i];
                dbuf[16 + mb + i] = d1[i];
                dbuf[32 + mb + i] = d2[i];
                dbuf[48 + mb + i] = d3[i];
            }
        }
        __syncthreads();
        // cw[b,k,s] += scalar * delta[s]
        float* cwrow = cw + (size_t)b * COLS + (size_t)k * SEQL;
        for (int s = lane; s < SEQL; s += 32)
            cwrow[s] += scalar * dbuf[s];
    }
}

// ---------------------------------------------------------------------------
extern "C" void kernel_launch(void* const* d_in, const int* in_sizes, int n_in,
                              void* d_out, int out_size, void* d_ws, size_t ws_size,
                              hipStream_t stream) {
    const int*   mask  = (const int*)d_in[0];    // [B,SEQ] int32
    const float* x     = (const float*)d_in[1];  // [B,SEQ,K,H] f32
    const float* cw_in = (const float*)d_in[2];  // [B,K,SEQ] f32
    float* out = (float*)d_out;                  // [B,K,H] f32

    float* cw_ws  = (float*)d_ws;                       // B*K*SEQ floats
    float* colmax = cw_ws + (size_t)BATCH * COLS;       // COLS floats
    float* colsum = colmax + COLS;                      // COLS floats

    const int ncw = BATCH * COLS;
    cw_copy_kernel<<<(ncw + 255) / 256, 256, 0, stream>>>(cw_in, cw_ws, ncw);

    for (int it = 0; it < 3; ++it) {
        col_stats_kernel<<<COLS, 256, 0, stream>>>(cw_ws, colmax, colsum);
        if (it < 2) {
            route_iter_kernel<false><<<BATCH * KNUM, 32, 0, stream>>>(
                x, mask, cw_ws, colmax, colsum, nullptr);
        } else {
            route_iter_kernel<true><<<BATCH * KNUM, 32, 0, stream>>>(
                x, mask, cw_ws, colmax, colsum, out);
        }
    }
}